// TripletLoss_33913061769800
// MI455X (gfx1250) — compile-verified
//
#include <hip/hip_runtime.h>
#include <hip/hip_bf16.h>
#include <stdint.h>

#define MARGIN  0.1f
#define NCOLS   4096
#define TPB     256
#define CHUNKS  (NCOLS / (TPB * 4))   // 4 float4 chunks per thread

typedef float v4f __attribute__((ext_vector_type(4)));
typedef int   v4i __attribute__((ext_vector_type(4)));

// Deterministic per-(row,col) uniform in [0,1): stand-in for threefry noise.
__device__ __forceinline__ float urand(unsigned row, unsigned col) {
  unsigned x = row * 0x9E3779B9u ^ col * 0x85EBCA6Bu ^ (42u * 0x27D4EB2Fu);
  x ^= x >> 16; x *= 0x7FEB352Du;
  x ^= x >> 15; x *= 0x846CA68Bu;
  x ^= x >> 16;
  return (float)(x >> 8) * 0x1p-24f;
}

__global__ __launch_bounds__(TPB) void triplet_row_kernel(
    const float*     __restrict__ scores,
    const int*       __restrict__ cls,
    const long long* __restrict__ sent_gt,
    float*           __restrict__ out_scores,  // d_out + 1 (copy of scores)
    float*           __restrict__ row_loss)    // d_ws, B floats
{
  __shared__ __align__(16) float smem[NCOLS];  // 16 KB staged score row
  __shared__ float redU[TPB];
  __shared__ float redS[TPB];
  __shared__ int   redI[TPB];

  const int b = blockIdx.x;
  const int t = threadIdx.x;
  const float* srow = scores + (size_t)b * NCOLS;
  const int*   crow = cls    + (size_t)b * NCOLS;

  // ---- CDNA5 async DMA: stage the score row global -> LDS (ASYNCcnt path) ----
#pragma unroll
  for (int c = 0; c < CHUNKS; ++c) {
    const int col = c * (TPB * 4) + t * 4;
    unsigned lds_off = (unsigned)(uint64_t)(uintptr_t)&smem[col]; // low 32b = LDS byte addr
    const float* g = srow + col;
    asm volatile("global_load_async_to_lds_b128 %0, %1, off"
                 :: "v"(lds_off), "v"(g) : "memory");
  }
  asm volatile("s_wait_asynccnt 0" ::: "memory");
  __syncthreads();

  const int   gt  = (int)sent_gt[b];
  const float pos = smem[gt];             // LDS broadcast of positive score

  float bU = -1.0f; int bI = 0; float bS = 0.0f;

#pragma unroll
  for (int c = 0; c < CHUNKS; ++c) {
    const int col = c * (TPB * 4) + t * 4;
    v4f s  = *(const v4f*)&smem[col];                               // ds_load_b128
    v4i cl = __builtin_nontemporal_load((const v4i*)(crow + col));  // NT b128 stream

    // Streaming copy of scores to output (out+1 is only 4B-aligned -> b32 NT stores)
    float* o = out_scores + (size_t)b * NCOLS + col;
    __builtin_nontemporal_store(s.x, o + 0);
    __builtin_nontemporal_store(s.y, o + 1);
    __builtin_nontemporal_store(s.z, o + 2);
    __builtin_nontemporal_store(s.w, o + 3);

#pragma unroll
    for (int j = 0; j < 4; ++j) {
      const int   n  = col + j;
      const float sv = s[j];
      const bool hard = (cl[j] != -1) && (n != gt) && (MARGIN + sv - pos > 0.0f);
      if (hard) {
        float u = urand((unsigned)b, (unsigned)n);
        if (u > bU || (u == bU && n < bI)) { bU = u; bI = n; bS = sv; }
      }
    }
  }

  // Block-wide argmax reduction (lowest index wins ties, matching jnp.argmax)
  redU[t] = bU; redI[t] = bI; redS[t] = bS;
  __syncthreads();
#pragma unroll
  for (int st = TPB / 2; st > 0; st >>= 1) {
    if (t < st) {
      float u2 = redU[t + st];
      if (u2 > redU[t] || (u2 == redU[t] && redI[t + st] < redI[t])) {
        redU[t] = u2; redI[t] = redI[t + st]; redS[t] = redS[t + st];
      }
    }
    __syncthreads();
  }

  if (t == 0) {
    float loss = 0.0f;
    if (redU[0] >= 0.0f)  // has at least one hard negative
      loss = fmaxf(MARGIN + redS[0] - pos, 0.0f);
    row_loss[b] = loss;
  }
}

// Deterministic final mean (avoids fp-atomic ordering nondeterminism).
__global__ __launch_bounds__(TPB) void mean_kernel(
    const float* __restrict__ row_loss, float* __restrict__ out, int nb)
{
  __shared__ float red[TPB];
  float s = 0.0f;
  for (int i = threadIdx.x; i < nb; i += TPB) s += row_loss[i];
  red[threadIdx.x] = s;
  __syncthreads();
#pragma unroll
  for (int st = TPB / 2; st > 0; st >>= 1) {
    if (threadIdx.x < st) red[threadIdx.x] += red[threadIdx.x + st];
    __syncthreads();
  }
  if (threadIdx.x == 0) out[0] = red[0] / (float)nb;
}

extern "C" void kernel_launch(void* const* d_in, const int* in_sizes, int n_in,
                              void* d_out, int out_size, void* d_ws, size_t ws_size,
                              hipStream_t stream) {
  const float*     scores  = (const float*)d_in[0];
  // d_in[1] (box, 512 MiB) is mathematically unused -> deliberately never read.
  const int*       cls     = (const int*)d_in[2];
  const long long* sent_gt = (const long long*)d_in[3];

  const int B = in_sizes[3];          // 8192 rows; NCOLS fixed at 4096 per reference

  float* out = (float*)d_out;         // out[0] = mean loss, out[1..] = scores copy
  float* ws  = (float*)d_ws;          // B per-row losses

  triplet_row_kernel<<<B, TPB, 0, stream>>>(scores, cls, sent_gt, out + 1, ws);
  mean_kernel<<<1, TPB, 0, stream>>>(ws, out, B);
}